// KGL_42176578847220
// MI455X (gfx1250) — compile-verified
//
#include <hip/hip_runtime.h>

// KGL kernel-generation layer for MI455X (gfx1250, wave32).
// One wave = one patch (8192 patches total), 8 waves / 256-thread block.
// Conv expressed as a padded 16x16x84 GEMM via V_WMMA_F32_16X16X4_F32.

typedef float v8f __attribute__((ext_vector_type(8)));
typedef float v2f __attribute__((ext_vector_type(2)));

#define ROWSTRIDE (768 * 9)          // floats per image row (W*C)
#define IMGSTRIDE (768 * 768 * 9)    // floats per image
#define KPAD 84                      // K=81 padded to multiple of 4
#define NPATCH 8192
#define WG1_TOTAL (NPATCH * 81)

__global__ __launch_bounds__(256) void kgl_kernel(
    const float* __restrict__ x,        // [8,768,768,9]
    const float* __restrict__ conv_w,   // [3,3,9,9]  flat: k*9+co, k=ky*27+kx*9+ci
    const float* __restrict__ conv_b,   // [9]
    const float* __restrict__ dense_w,  // [9,81]
    const float* __restrict__ dense_b,  // [81]
    float* __restrict__ out)            // wg1[8192*81] ++ wg2[8192*81]
{
    __shared__ float lds_A[8][16 * KPAD];   // per-wave padded im2col (43 KB)
    __shared__ float lds_B[KPAD * 16];      // padded conv weights  (5.3 KB)
    __shared__ float s_d[8][81];            // dense outputs per wave
    __shared__ float s_scale[8][9];         // wg2 column scales

    const int tid  = threadIdx.x;
    const int wave = tid >> 5;
    const int lane = tid & 31;
    const int r    = blockIdx.x * 8 + wave;     // flat patch row 0..8191

    // ---- stage zero-padded B = conv_w into LDS (once per block) ----
    for (int idx = tid; idx < KPAD * 16; idx += 256) {
        int k = idx >> 4, n = idx & 15;
        lds_B[idx] = (k < 81 && n < 9) ? conv_w[k * 9 + n] : 0.0f;
    }

    // ---- patch base pointer (r enumerates (b, prow, pcol) row-major) ----
    const int b_img = r >> 10;
    const int prow  = (r >> 5) & 31;
    const int pcol  = r & 31;
    const float* pbase = x + (size_t)b_img * IMGSTRIDE
                           + (size_t)(prow * 24) * ROWSTRIDE
                           + (size_t)(pcol * 24) * 9;

    // ---- zero + fill this wave's padded im2col A (16 x 84) ----
    float* A = &lds_A[wave][0];
    for (int idx = lane; idx < 16 * KPAD; idx += 32) A[idx] = 0.0f;
    for (int e = lane; e < 729; e += 32) {           // 9 pos * 81 taps
        int ci = e % 9;
        int t2 = e / 9;
        int fx = t2 % 9, fy = t2 / 9;                // footprint coords 0..8
        int oy = fy / 3, ky = fy % 3;
        int ox = fx / 3, kx = fx % 3;
        int ry = oy * 8 + ky, cx = ox * 8 + kx;      // rows/cols {0..2,8..10,16..18}
        A[(oy * 3 + ox) * KPAD + ky * 27 + kx * 9 + ci] =
            pbase[ry * ROWSTRIDE + cx * 9 + ci];
    }

    // ---- GAP: stream all 24*24 spatial positions, 9 channels each ----
    float sums[9];
#pragma unroll
    for (int c = 0; c < 9; ++c) sums[c] = 0.0f;
    for (int i = 0; i < 18; ++i) {                   // 18*32 = 576 positions
        int s = i * 32 + lane;
        int srow = s / 24, scol = s % 24;
        const float* p = pbase + srow * ROWSTRIDE + scol * 9;
#pragma unroll
        for (int c = 0; c < 9; ++c) sums[c] += p[c];
    }
#pragma unroll
    for (int c = 0; c < 9; ++c) {                    // wave32 butterfly reduce
        float v = sums[c];
        v += __shfl_xor(v, 16, 32);
        v += __shfl_xor(v, 8, 32);
        v += __shfl_xor(v, 4, 32);
        v += __shfl_xor(v, 2, 32);
        v += __shfl_xor(v, 1, 32);
        sums[c] = v * (1.0f / 576.0f);               // GAP mean, all lanes
    }

    __syncthreads();                                 // A and B staged

    // ---- conv as WMMA: D(16x16) = A(16x84) * B(84x16), 21 K-steps of 4 ----
    const int mn   = lane & 15;                      // M row (A) / N col (B,D)
    const int half = lane >> 4;
    v8f acc = {};
#pragma unroll
    for (int t = 0; t < 21; ++t) {
        int koff = 4 * t + 2 * half;                 // K = 2*half + vgpr
        v2f a, b;
        a.x = A[mn * KPAD + koff];
        a.y = A[mn * KPAD + koff + 1];
        b.x = lds_B[koff * 16 + mn];
        b.y = lds_B[(koff + 1) * 16 + mn];
        acc = __builtin_amdgcn_wmma_f32_16x16x4_f32(
            false, a, false, b, (short)0, acc, false, false);
    }

    // ---- add conv bias per out-channel (N = mn) ----
    float cb = (mn < 9) ? conv_b[mn] : 0.0f;
#pragma unroll
    for (int v = 0; v < 8; ++v) acc[v] += cb;

    // ---- wg1 clip-by-norm over 9 output positions per out-channel ----
    // lanes 0-15 hold p = 0..7 (M = v), lanes 16-31 hold p = 8 in acc[0]
    float ss;
    if (half == 0) {
        ss = 0.0f;
#pragma unroll
        for (int v = 0; v < 8; ++v) ss += acc[v] * acc[v];
    } else {
        ss = acc[0] * acc[0];
    }
    ss += __shfl_xor(ss, 16, 32);
    float scale1 = 1.0f / fmaxf(sqrtf(ss), 1.0f);

    float* out1 = out + (size_t)r * 81;
    if (mn < 9) {
        if (half == 0) {
#pragma unroll
            for (int v = 0; v < 8; ++v) out1[v * 9 + mn] = acc[v] * scale1;
        } else {
            out1[72 + mn] = acc[0] * scale1;         // p = 8 row
        }
    }

    // ---- wg2: dense 9 -> 81, then clip-by-norm per output column j ----
    for (int j = lane; j < 81; j += 32) {
        float d = dense_b[j];
#pragma unroll
        for (int i = 0; i < 9; ++i) d += sums[i] * dense_w[i * 81 + j];
        s_d[wave][j] = d;
    }
    __syncthreads();
    if (lane < 9) {
        float s2 = 0.0f;
#pragma unroll
        for (int i = 0; i < 9; ++i) {
            float dv = s_d[wave][i * 9 + lane];
            s2 += dv * dv;
        }
        s_scale[wave][lane] = 1.0f / fmaxf(sqrtf(s2), 1.0f);
    }
    __syncthreads();
    float* out2 = out + (size_t)WG1_TOTAL + (size_t)r * 81;
    for (int m = lane; m < 81; m += 32)
        out2[m] = s_d[wave][m] * s_scale[wave][m % 9];
}

extern "C" void kernel_launch(void* const* d_in, const int* in_sizes, int n_in,
                              void* d_out, int out_size, void* d_ws, size_t ws_size,
                              hipStream_t stream) {
    const float* x       = (const float*)d_in[0];
    const float* conv_w  = (const float*)d_in[1];
    const float* conv_b  = (const float*)d_in[2];
    const float* dense_w = (const float*)d_in[3];
    const float* dense_b = (const float*)d_in[4];
    float* out = (float*)d_out;
    // 8192 patches, one wave each, 8 waves per 256-thread block -> 1024 blocks
    kgl_kernel<<<NPATCH / 8, 256, 0, stream>>>(x, conv_w, conv_b,
                                               dense_w, dense_b, out);
}